// GraphCTH_NODE_52286931862126
// MI455X (gfx1250) — compile-verified
//
#include <hip/hip_runtime.h>
#include <hip/hip_bf16.h>
#include <math.h>

// Problem constants
#define BB 64
#define NN 307
#define TT 64
#define FF 3
#define HH 64
#define NP 320              // N padded to multiple of 32 (and of 4 for K-loop)
#define COLS (BB*HH)        // 4096 columns in node-major GEMM

typedef __attribute__((ext_vector_type(2))) float v2f;
typedef __attribute__((ext_vector_type(8))) float v8f;

#define WMMA_F32(a, b, c) \
    __builtin_amdgcn_wmma_f32_16x16x4_f32(false, (a), false, (b), (short)0, (c), false, false)

// ---------------------------------------------------------------------------
// Zero-pad A (307x307) into Ap (320x320)
// ---------------------------------------------------------------------------
__global__ void pad_A_kernel(const float* __restrict__ A, float* __restrict__ Ap) {
    int idx = blockIdx.x * 256 + threadIdx.x;
    if (idx >= NP * NP) return;
    int i = idx / NP, j = idx % NP;
    Ap[idx] = (i < NN && j < NN) ? A[i * NN + j] : 0.0f;
}

// ---------------------------------------------------------------------------
// fp32 WMMA GEMM: C[M x Ncols] = A[M x K] * B[K x Ncols], row-major.
// One wave computes a 32x64 tile with V_WMMA_F32_16X16X4_F32:
//   2 A-fragments (16x4 each) x 4 B-fragments (4x16 each) -> 8 accumulators.
// Per K-step: 2 x b64 + 8 x b32 loads feed 8 WMMAs (1.25 VMEM/WMMA).
// 32-bit element offsets keep addressing in saddr+voffset form.
// Grid sized so every wave owns a tile -> EXEC all-1s at every WMMA.
// K must be a multiple of 4; optional fused tanh epilogue.
// ---------------------------------------------------------------------------
__global__ void wmma_gemm_f32(const float* __restrict__ A, int lda,
                              const float* __restrict__ B, int ldb,
                              float* __restrict__ C, int ldc,
                              int K, int ctiles, int total_tiles, int apply_tanh)
{
    int wave = (blockIdx.x * blockDim.x + threadIdx.x) >> 5;
    int lane = threadIdx.x & 31;
    if (wave >= total_tiles) return;      // wave-uniform

    int mt = wave / ctiles;
    int ct = wave - mt * ctiles;
    int m0 = mt << 5;                     // 32 rows
    int c0 = ct << 6;                     // 64 cols

    int la = lane & 15;                   // lane within half-wave
    int kh = (lane >> 4) << 1;            // 0 for lanes 0-15, 2 for lanes 16-31

    v8f a00 = {}, a01 = {}, a02 = {}, a03 = {};
    v8f a10 = {}, a11 = {}, a12 = {}, a13 = {};

    // 32-bit element offsets (max 5.2MB buffers)
    int aoff0 = (m0 + la) * lda + kh;     // A[m0+la][kh+k..]
    int aoff1 = aoff0 + 16 * lda;         // A[m0+16+la][kh+k..]
    const float* bp = B + c0 + la;
    int boff  = kh * ldb;                 // B[kh+k][c0+la]
    int bstep = ldb << 2;

    for (int k = 0; k < K; k += 4) {
        // A fragments: 16x4 (MxK): lanes 0-15 hold K=k..k+1, lanes 16-31 K=k+2..k+3
        v2f af0 = *(const v2f*)(A + aoff0 + k);    // 8B aligned (k%4==0, lda%4==0)
        v2f af1 = *(const v2f*)(A + aoff1 + k);

        // B fragments: 4x16 (KxN): vgpr0 -> K=k+kh, vgpr1 -> K=k+kh+1
        v2f b0, b1, b2, b3;
        b0.x = bp[boff];      b0.y = bp[boff + ldb];
        b1.x = bp[boff + 16]; b1.y = bp[boff + ldb + 16];
        b2.x = bp[boff + 32]; b2.y = bp[boff + ldb + 32];
        b3.x = bp[boff + 48]; b3.y = bp[boff + ldb + 48];
        boff += bstep;

        a00 = WMMA_F32(af0, b0, a00);
        a01 = WMMA_F32(af0, b1, a01);
        a02 = WMMA_F32(af0, b2, a02);
        a03 = WMMA_F32(af0, b3, a03);
        a10 = WMMA_F32(af1, b0, a10);
        a11 = WMMA_F32(af1, b1, a11);
        a12 = WMMA_F32(af1, b2, a12);
        a13 = WMMA_F32(af1, b3, a13);
    }

    // C/D layout: vgpr r: lanes0-15 -> row m0+r, lanes16-31 -> row m0+8+r, col c0+la
    int rbase = m0 + ((lane >> 4) << 3);
    float* crow = C + (size_t)rbase * ldc + c0 + la;
    #pragma unroll
    for (int r = 0; r < 8; ++r) {
        float* cp = crow + (size_t)r * ldc;
        float x0 = a00[r], x1 = a01[r], x2 = a02[r], x3 = a03[r];
        if (apply_tanh) { x0 = tanhf(x0); x1 = tanhf(x1); x2 = tanhf(x2); x3 = tanhf(x3); }
        cp[0] = x0; cp[16] = x1; cp[32] = x2; cp[48] = x3;
    }
    crow += (size_t)16 * ldc;
    #pragma unroll
    for (int r = 0; r < 8; ++r) {
        float* cp = crow + (size_t)r * ldc;
        float x0 = a10[r], x1 = a11[r], x2 = a12[r], x3 = a13[r];
        if (apply_tanh) { x0 = tanhf(x0); x1 = tanhf(x1); x2 = tanhf(x2); x3 = tanhf(x3); }
        cp[0] = x0; cp[16] = x1; cp[32] = x2; cp[48] = x3;
    }
}

// ---------------------------------------------------------------------------
// Init: z0[n][b][:] = x_seq[b][n][0][:] @ W_enc + b_enc ; pad rows zero;
// pred[b][n][0] = z0 . W_dec + b_dec.  One wave (32 lanes, 2 elems/lane) per row.
// ---------------------------------------------------------------------------
__global__ void init_kernel(const float* __restrict__ x_seq,
                            const float* __restrict__ W_enc, const float* __restrict__ b_enc,
                            const float* __restrict__ W_dec, const float* __restrict__ b_dec,
                            float* __restrict__ z, float* __restrict__ pred)
{
    int row  = (blockIdx.x * blockDim.x + threadIdx.x) >> 5;   // 0 .. NP*BB-1
    int lane = threadIdx.x & 31;
    if (row >= NP * BB) return;
    int n = row >> 6, b = row & 63;
    int e0 = lane, e1 = lane + 32;
    float z0 = 0.0f, z1 = 0.0f;
    if (n < NN) {
        const float* xp = x_seq + ((size_t)(b * NN + n) * TT + 0) * FF;
        float x0 = xp[0], x1 = xp[1], x2 = xp[2];
        z0 = b_enc[e0] + x0 * W_enc[e0] + x1 * W_enc[HH + e0] + x2 * W_enc[2 * HH + e0];
        z1 = b_enc[e1] + x0 * W_enc[e1] + x1 * W_enc[HH + e1] + x2 * W_enc[2 * HH + e1];
    }
    size_t off = (size_t)row * HH;
    z[off + e0] = z0;
    z[off + e1] = z1;
    if (n < NN) {
        float p = z0 * W_dec[e0] + z1 * W_dec[e1];
        for (int o = 16; o; o >>= 1) p += __shfl_xor(p, o, 32);
        if (lane == 0) pred[(size_t)(b * NN + n) * TT + 0] = p + b_dec[0];
    }
}

// ---------------------------------------------------------------------------
// Fused per-step pointwise kernel. One wave per (n,b) row (2 H-elems / lane):
//   delta = LayerNorm(h2)*g + b ; z = 2z + delta ; z_obs = x_t @ W_obs + b_obs
//   gate  = sigmoid([z, z_obs] @ W_gate + b_gate)   (via per-wave LDS staging)
//   z    += gate * (z_obs - z) * mask ; pred[t] = z . W_dec + b_dec
// Grid covers exactly NN*BB rows -> barrier is uniform.
// ---------------------------------------------------------------------------
__global__ void step_update(const float* __restrict__ h2,
                            float* __restrict__ z,
                            const float* __restrict__ x_seq,
                            const float* __restrict__ obs_mask,
                            const float* __restrict__ ln_g, const float* __restrict__ ln_b,
                            const float* __restrict__ W_obs, const float* __restrict__ b_obs,
                            const float* __restrict__ W_gate, const float* __restrict__ b_gate,
                            const float* __restrict__ W_dec, const float* __restrict__ b_dec,
                            float* __restrict__ pred, int t)
{
    __shared__ float cat[4][2 * HH];
    int wid  = threadIdx.x >> 5;
    int lane = threadIdx.x & 31;
    int row  = blockIdx.x * 4 + wid;           // < NN*BB by construction
    int n = row >> 6, b = row & 63;
    int e0 = lane, e1 = lane + 32;
    size_t off = (size_t)row * HH;

    // LayerNorm over 64 via wave32 shuffle reductions
    float h0 = h2[off + e0], h1 = h2[off + e1];
    float s = h0 + h1;
    for (int o = 16; o; o >>= 1) s += __shfl_xor(s, o, 32);
    float mu = s * (1.0f / 64.0f);
    float d0 = h0 - mu, d1 = h1 - mu;
    float v = d0 * d0 + d1 * d1;
    for (int o = 16; o; o >>= 1) v += __shfl_xor(v, o, 32);
    float rinv = rsqrtf(v * (1.0f / 64.0f) + 1e-5f);
    float delta0 = d0 * rinv * ln_g[e0] + ln_b[e0];
    float delta1 = d1 * rinv * ln_g[e1] + ln_b[e1];

    // z = z + (z + delta)
    float z0 = 2.0f * z[off + e0] + delta0;
    float z1 = 2.0f * z[off + e1] + delta1;

    // z_obs = x_t @ W_obs + b_obs   (F = 3)
    const float* xp = x_seq + ((size_t)(b * NN + n) * TT + t) * FF;
    float x0 = xp[0], x1 = xp[1], x2 = xp[2];
    float zb0 = b_obs[e0] + x0 * W_obs[e0] + x1 * W_obs[HH + e0] + x2 * W_obs[2 * HH + e0];
    float zb1 = b_obs[e1] + x0 * W_obs[e1] + x1 * W_obs[HH + e1] + x2 * W_obs[2 * HH + e1];

    // gate = sigmoid(concat(z, z_obs) @ W_gate + b_gate)
    cat[wid][e0] = z0;  cat[wid][e1] = z1;
    cat[wid][HH + e0] = zb0;  cat[wid][HH + e1] = zb1;
    __syncthreads();
    float g0 = b_gate[e0], g1 = b_gate[e1];
    const float* cp = cat[wid];
    #pragma unroll 4
    for (int i = 0; i < 2 * HH; ++i) {
        float c = cp[i];
        g0 += c * W_gate[i * HH + e0];
        g1 += c * W_gate[i * HH + e1];
    }
    g0 = 1.0f / (1.0f + __expf(-g0));
    g1 = 1.0f / (1.0f + __expf(-g1));

    float m = obs_mask[b * NN + n];
    float zn0 = z0 + g0 * (zb0 - z0) * m;
    float zn1 = z1 + g1 * (zb1 - z1) * m;
    z[off + e0] = zn0;
    z[off + e1] = zn1;

    float p = zn0 * W_dec[e0] + zn1 * W_dec[e1];
    for (int o = 16; o; o >>= 1) p += __shfl_xor(p, o, 32);
    if (lane == 0) pred[(size_t)(b * NN + n) * TT + t] = p + b_dec[0];
}

// ---------------------------------------------------------------------------
extern "C" void kernel_launch(void* const* d_in, const int* in_sizes, int n_in,
                              void* d_out, int out_size, void* d_ws, size_t ws_size,
                              hipStream_t stream) {
    (void)in_sizes; (void)n_in; (void)out_size; (void)ws_size;
    const float* x_seq    = (const float*)d_in[0];
    const float* obs_mask = (const float*)d_in[1];
    const float* A        = (const float*)d_in[2];
    const float* W_enc    = (const float*)d_in[3];
    const float* b_enc    = (const float*)d_in[4];
    const float* W_gc1    = (const float*)d_in[5];
    const float* W_gc2    = (const float*)d_in[6];
    const float* ln_g     = (const float*)d_in[7];
    const float* ln_b     = (const float*)d_in[8];
    const float* W_obs    = (const float*)d_in[9];
    const float* b_obs    = (const float*)d_in[10];
    const float* W_gate   = (const float*)d_in[11];
    const float* b_gate   = (const float*)d_in[12];
    const float* W_dec    = (const float*)d_in[13];
    const float* b_dec    = (const float*)d_in[14];
    float* out = (float*)d_out;

    // Workspace layout (all 256B-aligned slices)
    char* ws = (char*)d_ws;
    const size_t szA = (size_t)NP * NP * sizeof(float);          // 400 KB
    const size_t szM = (size_t)NP * COLS * sizeof(float);        // 5.24 MB
    float* Ap = (float*)ws;
    float* zb = (float*)(ws + szA);
    float* t1 = (float*)(ws + szA + szM);
    float* t2 = (float*)(ws + szA + 2 * szM);

    pad_A_kernel<<<(NP * NP + 255) / 256, 256, 0, stream>>>(A, Ap);
    init_kernel<<<(NP * BB) / 4, 128, 0, stream>>>(x_seq, W_enc, b_enc, W_dec, b_dec, zb, out);

    // GEMM1: (320x320) x (320x4096); 10 * 64 = 640 tile-waves -> 80 blocks of 8 waves
    // GEMM2: rows = 320*64 = 20480 -> 640 M-tiles of 32, 1 col-group -> 80 blocks
    const int g1_tiles = (NP / 32) * (COLS / 64);        // 640
    const int g2_tiles = (NP * BB) / 32;                 // 640

    for (int t = 1; t < TT; ++t) {
        wmma_gemm_f32<<<80, 256, 0, stream>>>(Ap, NP, zb, COLS, t1, COLS, NP, COLS / 64, g1_tiles, 0);
        wmma_gemm_f32<<<80, 256, 0, stream>>>(t1, HH, W_gc1, HH, t2, HH, HH, 1, g2_tiles, 1);
        wmma_gemm_f32<<<80, 256, 0, stream>>>(Ap, NP, t2, COLS, t1, COLS, NP, COLS / 64, g1_tiles, 0);
        wmma_gemm_f32<<<80, 256, 0, stream>>>(t1, HH, W_gc2, HH, t2, HH, HH, 1, g2_tiles, 1);
        step_update<<<(NN * BB) / 4, 128, 0, stream>>>(t2, zb, x_seq, obs_mask, ln_g, ln_b,
                                                       W_obs, b_obs, W_gate, b_gate,
                                                       W_dec, b_dec, out, t);
    }
}